// GATLayer_6081673691288
// MI455X (gfx1250) — compile-verified
//
#include <hip/hip_runtime.h>
#include <math.h>

typedef __attribute__((ext_vector_type(2))) float v2f;
typedef __attribute__((ext_vector_type(4))) float v4f;
typedef __attribute__((ext_vector_type(8))) float v8f;

#define B_DIM   8
#define N_DIM   4096
#define IN_DIM  256
#define OUT_DIM 64
#define NROWS   (B_DIM * N_DIM)        // 32768 rows, batch is contiguous -> one big GEMM
#define WAVES_PER_WG 8
#define WG_THREADS   (WAVES_PER_WG * 32)

// ---------------------------------------------------------------------------
// Kernel 1: h[32768,64] = x[32768,256] @ W_fc[256,64] + b_fc, via
// V_WMMA_F32_16X16X4_F32. W_fc is staged in LDS pre-swizzled into the exact
// B-fragment register layout (lane l: rows kbase+(l>>4)*2 .. +1, col n0+(l&15))
// so each fragment load is a single 8B LDS read per lane.
// Each wave: one 16-row x 64-col strip = 4 fp32 accumulators, 64 K-steps.
// ---------------------------------------------------------------------------
__global__ __launch_bounds__(WG_THREADS) void gemm_h_kernel(
    const float* __restrict__ x, const float* __restrict__ W,
    const float* __restrict__ bfc, float* __restrict__ h)
{
    // 256 fragments (64 ksteps * 4 ntiles) * 32 lanes * 2 floats = 64 KB
    __shared__ float wfrag[256 * 64];

    const int tid = threadIdx.x;

    // Cooperative swizzled fill of W fragments.
    for (int idx = tid; idx < 256 * 64; idx += WG_THREADS) {
        int f     = idx >> 6;          // fragment id = kstep*4 + ntile
        int r     = idx & 63;
        int lane  = r >> 1;
        int t     = r & 1;
        int kstep = f >> 2;
        int ntile = f & 3;
        int k = kstep * 4 + ((lane >> 4) << 1) + t;   // B-frag row (K)
        int n = ntile * 16 + (lane & 15);             // B-frag col (N)
        wfrag[idx] = W[k * OUT_DIM + n];
    }
    __syncthreads();

    const int wave = tid >> 5;
    const int lane = tid & 31;
    const int m0   = (blockIdx.x * WAVES_PER_WG + wave) * 16;

    // A-fragment source: lane l -> row m0+(l&15), cols kbase+(l>>4)*2, +1  (b64 load)
    const float* aptr = x + (size_t)(m0 + (lane & 15)) * IN_DIM + ((lane >> 4) << 1);

    v8f acc0 = {}; v8f acc1 = {}; v8f acc2 = {}; v8f acc3 = {};

    #pragma unroll 8
    for (int kstep = 0; kstep < IN_DIM / 4; ++kstep) {
        v2f a = *(const v2f*)(aptr + kstep * 4);
        const float* bb = &wfrag[(kstep * 4) * 64 + lane * 2];
        v2f b0 = *(const v2f*)(bb + 0 * 64);
        v2f b1 = *(const v2f*)(bb + 1 * 64);
        v2f b2 = *(const v2f*)(bb + 2 * 64);
        v2f b3 = *(const v2f*)(bb + 3 * 64);
        acc0 = __builtin_amdgcn_wmma_f32_16x16x4_f32(false, a, false, b0, (short)0, acc0, false, false);
        acc1 = __builtin_amdgcn_wmma_f32_16x16x4_f32(false, a, false, b1, (short)0, acc1, false, false);
        acc2 = __builtin_amdgcn_wmma_f32_16x16x4_f32(false, a, false, b2, (short)0, acc2, false, false);
        acc3 = __builtin_amdgcn_wmma_f32_16x16x4_f32(false, a, false, b3, (short)0, acc3, false, false);
    }

    // Bias (one scalar per lane per n-tile; same across the 8 C rows).
    const int   col  = lane & 15;
    const float bs0  = bfc[ 0 + col];
    const float bs1  = bfc[16 + col];
    const float bs2  = bfc[32 + col];
    const float bs3  = bfc[48 + col];

    // C/D layout: element r of the v8f = row m0 + r + (lane>>4)*8, col = ntile*16 + (lane&15)
    float* hrow = h + (size_t)(m0 + ((lane >> 4) << 3)) * OUT_DIM + col;
    #pragma unroll
    for (int r = 0; r < 8; ++r) {
        float* hp = hrow + r * OUT_DIM;
        hp[ 0] = acc0[r] + bs0;
        hp[16] = acc1[r] + bs1;
        hp[32] = acc2[r] + bs2;
        hp[48] = acc3[r] + bs3;
    }
}

// ---------------------------------------------------------------------------
// Kernel 2: per batch b (one block each):
//   s2[j] = h[b,j,:] . a2          (a2 = w_attn[64:128])
//   w     = softmax_j(s2)          (s1 and b_attn cancel by shift-invariance)
//   c[b,:] = sum_j w[j] * h[b,j,:]
// ---------------------------------------------------------------------------
__global__ __launch_bounds__(256) void attn_reduce_kernel(
    const float* __restrict__ h, const float* __restrict__ w_attn,
    float* __restrict__ c)
{
    __shared__ float a2s[OUT_DIM];
    __shared__ float s2s[N_DIM];     // 16 KB
    __shared__ float red[256];

    const int tid = threadIdx.x;
    const int b   = blockIdx.x;
    const float* hb = h + (size_t)b * N_DIM * OUT_DIM;

    if (tid < OUT_DIM) a2s[tid] = w_attn[OUT_DIM + tid];
    __syncthreads();

    // Pass 1: s2 and block max
    float lmax = -INFINITY;
    for (int j = tid; j < N_DIM; j += 256) {
        const float* hr = hb + (size_t)j * OUT_DIM;
        float dot = 0.f;
        #pragma unroll
        for (int o = 0; o < OUT_DIM; o += 4) {
            v4f hv = *(const v4f*)(hr + o);
            dot += hv[0]*a2s[o] + hv[1]*a2s[o+1] + hv[2]*a2s[o+2] + hv[3]*a2s[o+3];
        }
        s2s[j] = dot;
        lmax = fmaxf(lmax, dot);
    }
    red[tid] = lmax; __syncthreads();
    for (int s = 128; s > 0; s >>= 1) {
        if (tid < s) red[tid] = fmaxf(red[tid], red[tid + s]);
        __syncthreads();
    }
    const float gmax = red[0];
    __syncthreads();

    // Pass 2: exp and block sum
    float lsum = 0.f;
    for (int j = tid; j < N_DIM; j += 256) {
        float w = __expf(s2s[j] - gmax);
        s2s[j] = w;
        lsum += w;
    }
    red[tid] = lsum; __syncthreads();
    for (int s = 128; s > 0; s >>= 1) {
        if (tid < s) red[tid] += red[tid + s];
        __syncthreads();
    }
    const float inv = 1.0f / red[0];
    __syncthreads();

    // Pass 3: weighted column sums. 256 threads = 4 j-groups x 64 columns.
    const int o = tid & 63;
    const int g = tid >> 6;
    float accv = 0.f;
    for (int j = g; j < N_DIM; j += 4)
        accv += s2s[j] * hb[(size_t)j * OUT_DIM + o];
    red[tid] = accv; __syncthreads();
    if (tid < OUT_DIM)
        c[b * OUT_DIM + tid] =
            (red[tid] + red[tid + 64] + red[tid + 128] + red[tid + 192]) * inv;
}

// ---------------------------------------------------------------------------
// Kernel 3: out[b,i,:] = c[b,:]  (float4 coalesced broadcast, 8.4 MB stores)
// ---------------------------------------------------------------------------
__global__ __launch_bounds__(256) void broadcast_kernel(
    const float* __restrict__ c, float* __restrict__ out)
{
    int idx = blockIdx.x * blockDim.x + threadIdx.x;   // over 8*4096*16 float4s
    int b   = idx >> 16;                               // 4096*16 per batch
    int o4  = idx & 15;
    ((v4f*)out)[idx] = ((const v4f*)c)[b * 16 + o4];
}

// ---------------------------------------------------------------------------
extern "C" void kernel_launch(void* const* d_in, const int* in_sizes, int n_in,
                              void* d_out, int out_size, void* d_ws, size_t ws_size,
                              hipStream_t stream)
{
    const float* x      = (const float*)d_in[0];
    const float* W_fc   = (const float*)d_in[1];
    const float* b_fc   = (const float*)d_in[2];
    const float* w_attn = (const float*)d_in[3];
    // d_in[4] = b_attn: cancels exactly under softmax shift-invariance.

    float* h = (float*)d_ws;                      // 32768*64 floats = 8 MB
    float* c = h + (size_t)NROWS * OUT_DIM;       // 8*64 floats

    gemm_h_kernel<<<NROWS / (WAVES_PER_WG * 16), WG_THREADS, 0, stream>>>(x, W_fc, b_fc, h);
    attn_reduce_kernel<<<B_DIM, 256, 0, stream>>>(h, w_attn, c);
    broadcast_kernel<<<(B_DIM * N_DIM * OUT_DIM / 4) / 256, 256, 0, stream>>>(c, (float*)d_out);
}